// GCN_57578331571013
// MI455X (gfx1250) — compile-verified
//
#include <hip/hip_runtime.h>
#include <hip/hip_bf16.h>
#include <math.h>

typedef __attribute__((ext_vector_type(2))) float v2f;
typedef __attribute__((ext_vector_type(8))) float v8f;

// ---------------------------------------------------------------------------
// Zero-fill a float region (workspace accumulators are poisoned by harness).
// ---------------------------------------------------------------------------
__global__ void zero_kernel(float* __restrict__ p, long n) {
    long i = (long)blockIdx.x * blockDim.x + threadIdx.x;
    if (i < n) p[i] = 0.0f;
}

// ---------------------------------------------------------------------------
// Layer-1 dense precompute with fp32 WMMA (K=16, N=16):
//   P = X @ W, Q = X @ B, R = X @ ROOT
// One wave per 16-node tile; K consumed in chunks of 4 via
// V_WMMA_F32_16X16X4_F32 (3 accumulators x 4 chunks = 12 WMMAs/wave).
//
// No load predication: addresses are clamped instead. A padded A-row only
// pollutes the matching (never-stored) output row, so garbage is harmless.
// Stores use a wave-uniform full-tile fast path (straight-line stores with
// immediate offsets); the guarded path runs only for a partial tail tile.
//
// VGPR layouts (ISA 7.12.2):
//  A 16x4 f32 : lanes 0-15 -> rows, comps {K=kb,kb+1}; lanes 16-31 -> {kb+2,kb+3}
//  B 4x16 f32 : lane n=lane&15 -> col; lanes<16 comps {K=kb,kb+1}; hi {kb+2,kb+3}
//  C 16x16 f32: lane n=lane&15 -> col; comp v -> row v + (lane>=16 ? 8 : 0)
// ---------------------------------------------------------------------------
__global__ void pq1_wmma_kernel(const float* __restrict__ X,     // [n_nodes,16]
                                const float* __restrict__ W,     // [16,16]
                                const float* __restrict__ B,     // [16,16]
                                const float* __restrict__ ROOT,  // [16,16]
                                float* __restrict__ P,
                                float* __restrict__ Q,
                                float* __restrict__ R,
                                int n_nodes) {
    const int gtid = blockIdx.x * blockDim.x + threadIdx.x;
    const int wave = gtid >> 5;                 // one wave per 16-node tile
    const int lane = threadIdx.x & 31;
    const int row0 = wave * 16;
    if (row0 >= n_nodes) return;                // wave-uniform exit

    const int  m  = lane & 15;                  // output column
    const bool hi = lane >= 16;
    int arow = row0 + m;                        // A-matrix row for this lane
    arow = (arow < n_nodes) ? arow : (n_nodes - 1);   // clamp, never stored if padded
    const float2* xrow = (const float2*)(X + (long)arow * 16);

    v8f accP = {}, accQ = {}, accR = {};

    #pragma unroll
    for (int c = 0; c < 4; ++c) {
        const int kb = c * 4 + (hi ? 2 : 0);    // K base for this lane group

        const float2 av = xrow[kb >> 1];        // global_load_b64, unconditional
        v2f a;  a.x = av.x;  a.y = av.y;

        v2f bw, bb, br;
        bw.x = W[kb * 16 + m];        bw.y = W[(kb + 1) * 16 + m];
        bb.x = B[kb * 16 + m];        bb.y = B[(kb + 1) * 16 + m];
        br.x = ROOT[kb * 16 + m];     br.y = ROOT[(kb + 1) * 16 + m];

        accP = __builtin_amdgcn_wmma_f32_16x16x4_f32(false, a, false, bw,
                                                     (short)0, accP, false, false);
        accQ = __builtin_amdgcn_wmma_f32_16x16x4_f32(false, a, false, bb,
                                                     (short)0, accQ, false, false);
        accR = __builtin_amdgcn_wmma_f32_16x16x4_f32(false, a, false, br,
                                                     (short)0, accR, false, false);
    }

    const long obase = (long)(row0 + (hi ? 8 : 0)) * 16 + m;
    if (row0 + 16 <= n_nodes) {
        // Full tile (wave-uniform): straight-line stores, immediate offsets.
        float* pp = P + obase;
        float* qq = Q + obase;
        float* rr = R + obase;
        #pragma unroll
        for (int v = 0; v < 8; ++v) {
            pp[v * 16] = accP[v];
            qq[v * 16] = accQ[v];
            rr[v * 16] = accR[v];
        }
    } else {
        // Partial tail tile: per-row guard.
        #pragma unroll
        for (int v = 0; v < 8; ++v) {
            const int orow = row0 + v + (hi ? 8 : 0);
            if (orow < n_nodes) {
                P[(long)orow * 16 + m] = accP[v];
                Q[(long)orow * 16 + m] = accQ[v];
                R[(long)orow * 16 + m] = accR[v];
            }
        }
    }
}

// ---------------------------------------------------------------------------
// Layer-2 dense precompute (K=16, fout=8) with column packing:
// the B operand is the packed 16x16 matrix [W2 | B2], so ONE accumulator
// yields p2 in columns 0-7 and q2 in columns 8-15 (no wasted lanes).
// root2 is replicated into both halves (via m&7) and stored from the low
// half only. 2 accumulators x 4 chunks = 8 WMMAs/wave.
// ---------------------------------------------------------------------------
__global__ void pq2_wmma_kernel(const float* __restrict__ H,    // [n_nodes,16]
                                const float* W,                 // [16,8]
                                const float* B,                 // [16,8]
                                const float* __restrict__ ROOT, // [16,8]
                                float* P,                       // [n_nodes,8]
                                float* Q,                       // [n_nodes,8]
                                float* __restrict__ R,          // [n_nodes,8]
                                int n_nodes) {
    const int gtid = blockIdx.x * blockDim.x + threadIdx.x;
    const int wave = gtid >> 5;
    const int lane = threadIdx.x & 31;
    const int row0 = wave * 16;
    if (row0 >= n_nodes) return;

    const int  m  = lane & 15;
    const int  mm = m & 7;
    const bool lo = (m < 8);
    const bool hi = lane >= 16;
    int arow = row0 + m;
    arow = (arow < n_nodes) ? arow : (n_nodes - 1);
    const float2* hrow = (const float2*)(H + (long)arow * 16);

    const float* WB = lo ? W : B;               // packed B-operand [W2 | B2]

    v8f accPQ = {}, accR = {};

    #pragma unroll
    for (int c = 0; c < 4; ++c) {
        const int kb = c * 4 + (hi ? 2 : 0);

        const float2 av = hrow[kb >> 1];
        v2f a;  a.x = av.x;  a.y = av.y;

        v2f bwb, brr;
        bwb.x = WB[kb * 8 + mm];       bwb.y = WB[(kb + 1) * 8 + mm];
        brr.x = ROOT[kb * 8 + mm];     brr.y = ROOT[(kb + 1) * 8 + mm];

        accPQ = __builtin_amdgcn_wmma_f32_16x16x4_f32(false, a, false, bwb,
                                                      (short)0, accPQ, false, false);
        accR  = __builtin_amdgcn_wmma_f32_16x16x4_f32(false, a, false, brr,
                                                      (short)0, accR, false, false);
    }

    float* PQ = lo ? P : Q;                     // cols 0-7 -> P, cols 8-15 -> Q
    const long obase = (long)(row0 + (hi ? 8 : 0)) * 8 + mm;
    if (row0 + 16 <= n_nodes) {
        // Full tile (wave-uniform): straight-line stores. All 32 lanes store
        // accPQ; only the low-column half stores R (single divergent mask).
        float* pq = PQ + obase;
        #pragma unroll
        for (int v = 0; v < 8; ++v) pq[v * 8] = accPQ[v];
        if (lo) {
            float* rr = R + obase;
            #pragma unroll
            for (int v = 0; v < 8; ++v) rr[v * 8] = accR[v];
        }
    } else {
        #pragma unroll
        for (int v = 0; v < 8; ++v) {
            const int orow = row0 + v + (hi ? 8 : 0);
            if (orow < n_nodes) {
                PQ[(long)orow * 8 + mm] = accPQ[v];
                if (lo) R[(long)orow * 8 + mm] = accR[v];
            }
        }
    }
}

// ---------------------------------------------------------------------------
// Edge scatter:  agg[dst] += a_e * p[src] + q[src];  cnt[dst] += 1 (opt).
// One thread per edge; float4 gathers (p/q/agg are L2-resident: ~3.2 MB each).
// ---------------------------------------------------------------------------
template <int FOUT, bool COUNT>
__global__ void edge_kernel(const int* __restrict__ ei,      // [2, n_edges]
                            const float* __restrict__ ea,    // [n_edges]
                            const float* __restrict__ p,     // [n_nodes,FOUT]
                            const float* __restrict__ q,     // [n_nodes,FOUT]
                            float* __restrict__ agg,         // [n_nodes,FOUT]
                            float* __restrict__ cnt,         // [n_nodes]
                            int n_edges) {
    const int e = blockIdx.x * blockDim.x + threadIdx.x;
    if (e >= n_edges) return;
    const int s = ei[e];
    const int d = ei[n_edges + e];
    const float a = ea[e];

    const float4* ps = (const float4*)(p + (long)s * FOUT);
    const float4* qs = (const float4*)(q + (long)s * FOUT);
    float* ad = agg + (long)d * FOUT;

    #pragma unroll
    for (int j = 0; j < FOUT / 4; ++j) {
        const float4 pv = ps[j];
        const float4 qv = qs[j];
        atomicAdd(ad + 4 * j + 0, fmaf(a, pv.x, qv.x));
        atomicAdd(ad + 4 * j + 1, fmaf(a, pv.y, qv.y));
        atomicAdd(ad + 4 * j + 2, fmaf(a, pv.z, qv.z));
        atomicAdd(ad + 4 * j + 3, fmaf(a, pv.w, qv.w));
    }
    if (COUNT) atomicAdd(cnt + d, 1.0f);
}

// ---------------------------------------------------------------------------
// Node update layer 1:  h = relu(agg/max(cnt,1) + x@root1 + bias1)
// One thread per (node, feature).
// ---------------------------------------------------------------------------
__global__ void node1_kernel(const float* __restrict__ agg,
                             const float* __restrict__ cnt,
                             const float* __restrict__ r1,
                             const float* __restrict__ bias1,
                             float* __restrict__ h, int n_nodes) {
    const long i = (long)blockIdx.x * blockDim.x + threadIdx.x;
    if (i >= (long)n_nodes * 16) return;
    const int v = (int)(i >> 4);
    const int j = (int)(i & 15);
    const float c = fmaxf(cnt[v], 1.0f);
    const float val = agg[i] / c + r1[i] + bias1[j];
    h[i] = fmaxf(val, 0.0f);
}

// ---------------------------------------------------------------------------
// Node update layer 2 + global mean pool accumulation.
// ---------------------------------------------------------------------------
__global__ void node2_pool_kernel(const float* __restrict__ agg2,
                                  const float* __restrict__ cnt,
                                  const float* __restrict__ r2,
                                  const float* __restrict__ bias2,
                                  const int* __restrict__ batch,
                                  float* __restrict__ pooled,
                                  float* __restrict__ gcnt, int n_nodes) {
    const int v = blockIdx.x * blockDim.x + threadIdx.x;
    if (v >= n_nodes) return;
    const float c = fmaxf(cnt[v], 1.0f);
    const int g = batch[v];
    #pragma unroll
    for (int j = 0; j < 8; ++j) {
        const float val = agg2[(long)v * 8 + j] / c + r2[(long)v * 8 + j] + bias2[j];
        atomicAdd(pooled + (long)g * 8 + j, val);
    }
    atomicAdd(gcnt + g, 1.0f);
}

// ---------------------------------------------------------------------------
// Final: per-graph mean + log_softmax over 8 classes. One thread per graph.
// ---------------------------------------------------------------------------
__global__ void final_kernel(const float* __restrict__ pooled,
                             const float* __restrict__ gcnt,
                             float* __restrict__ out, int n_graphs) {
    const int g = blockIdx.x * blockDim.x + threadIdx.x;
    if (g >= n_graphs) return;
    const float c = fmaxf(gcnt[g], 1.0f);
    float vals[8];
    float mx = -INFINITY;
    #pragma unroll
    for (int j = 0; j < 8; ++j) {
        vals[j] = pooled[(long)g * 8 + j] / c;
        mx = fmaxf(mx, vals[j]);
    }
    float s = 0.0f;
    #pragma unroll
    for (int j = 0; j < 8; ++j) s += expf(vals[j] - mx);
    const float lse = mx + logf(s);
    #pragma unroll
    for (int j = 0; j < 8; ++j) out[(long)g * 8 + j] = vals[j] - lse;
}

// ---------------------------------------------------------------------------
extern "C" void kernel_launch(void* const* d_in, const int* in_sizes, int n_in,
                              void* d_out, int out_size, void* d_ws, size_t ws_size,
                              hipStream_t stream) {
    const float* x      = (const float*)d_in[0];
    const int*   ei     = (const int*)  d_in[1];
    const float* ea     = (const float*)d_in[2];
    const int*   batch  = (const int*)  d_in[3];
    const float* lin1_w = (const float*)d_in[4];
    const float* lin1_b = (const float*)d_in[5];
    const float* root1  = (const float*)d_in[6];
    const float* bias1  = (const float*)d_in[7];
    const float* lin2_w = (const float*)d_in[8];
    const float* lin2_b = (const float*)d_in[9];
    const float* root2  = (const float*)d_in[10];
    const float* bias2  = (const float*)d_in[11];

    const int n_nodes  = in_sizes[0] / 16;
    const int n_edges  = in_sizes[2];
    const int n_graphs = out_size / 8;

    // Workspace carve-up (floats).
    float* ws = (float*)d_ws;
    const size_t N16 = (size_t)n_nodes * 16;
    const size_t N8  = (size_t)n_nodes * 8;
    float* p1   = ws;  ws += N16;
    float* q1   = ws;  ws += N16;
    float* r1   = ws;  ws += N16;
    float* h    = ws;  ws += N16;
    float* p2   = ws;  ws += N8;
    float* q2   = ws;  ws += N8;
    float* r2   = ws;  ws += N8;
    // Accumulators: contiguous so one zero-fill covers them all.
    float* agg1   = ws;  ws += N16;
    float* cnt    = ws;  ws += n_nodes;
    float* agg2   = ws;  ws += N8;
    float* pooled = ws;  ws += (size_t)n_graphs * 8;
    float* gcnt   = ws;  ws += n_graphs;
    const long nzero = (long)(N16 + n_nodes + N8 + (size_t)n_graphs * 9);

    const int BLK = 256;

    // 0) Zero the accumulators.
    zero_kernel<<<(int)((nzero + BLK - 1) / BLK), BLK, 0, stream>>>(agg1, nzero);

    // 1) Layer-1 dense precompute via fp32 WMMA: p1=x@W1, q1=x@B1, r1=x@root1.
    const int tiles = (n_nodes + 15) / 16;
    const int pq_threads = tiles * 32;
    pq1_wmma_kernel<<<(pq_threads + BLK - 1) / BLK, BLK, 0, stream>>>(
        x, lin1_w, lin1_b, root1, p1, q1, r1, n_nodes);

    // 2) Layer-1 edge scatter (also builds in-degree counts).
    edge_kernel<16, true><<<(n_edges + BLK - 1) / BLK, BLK, 0, stream>>>(
        ei, ea, p1, q1, agg1, cnt, n_edges);

    // 3) Layer-1 node update + ReLU -> h.
    const long n1 = (long)n_nodes * 16;
    node1_kernel<<<(int)((n1 + BLK - 1) / BLK), BLK, 0, stream>>>(
        agg1, cnt, r1, bias1, h, n_nodes);

    // 4) Layer-2 dense precompute via fp32 WMMA with [W2|B2] column packing.
    pq2_wmma_kernel<<<(pq_threads + BLK - 1) / BLK, BLK, 0, stream>>>(
        h, lin2_w, lin2_b, root2, p2, q2, r2, n_nodes);

    // 5) Layer-2 edge scatter (counts already built).
    edge_kernel<8, false><<<(n_edges + BLK - 1) / BLK, BLK, 0, stream>>>(
        ei, ea, p2, q2, agg2, cnt, n_edges);

    // 6) Layer-2 node update + per-graph pooling atomics.
    node2_pool_kernel<<<(n_nodes + BLK - 1) / BLK, BLK, 0, stream>>>(
        agg2, cnt, r2, bias2, batch, pooled, gcnt, n_nodes);

    // 7) Per-graph mean + log_softmax.
    final_kernel<<<(n_graphs + BLK - 1) / BLK, BLK, 0, stream>>>(
        pooled, gcnt, (float*)d_out, n_graphs);
}